// layer_with_lg_1_26113401160399
// MI455X (gfx1250) — compile-verified
//
#include <hip/hip_runtime.h>
#include <hip/hip_bf16.h>

// ---------------------------------------------------------------------------
// MI455X (gfx1250) implementation.
// All contractions use V_WMMA_F32_16X16X4_F32 (native f32 matrix path):
// inputs/outputs are f32 and the problem sits near the 23.3 TB/s bandwidth
// roofline (W/WL stream = 670 MB, passthrough copies = ~1.6 GB of traffic),
// so keeping full f32 precision costs nothing vs. the memory floor.
// ---------------------------------------------------------------------------

typedef float v2f __attribute__((ext_vector_type(2)));
typedef float v8f __attribute__((ext_vector_type(8)));

#if defined(__HIP_DEVICE_COMPILE__) && \
    !__has_builtin(__builtin_amdgcn_wmma_f32_16x16x4_f32)
#error "missing __builtin_amdgcn_wmma_f32_16x16x4_f32 on this toolchain"
#endif

__device__ __forceinline__ v8f wmma16x16x4(v2f a, v2f b, v8f c) {
  // (neg_a, A, neg_b, B, c_mod, C, reuse_a, reuse_b)
  return __builtin_amdgcn_wmma_f32_16x16x4_f32(false, a, false, b, (short)0, c,
                                               false, false);
}

// ---------------------------------------------------------------------------
// gop: Out[b, j*32+f, n] = sum_m W[b, n, m, j] * X[b, f, m]   (square: M==N)
// One workgroup = one 16-wide n tile, all J=4 j's and both f-halves (8 waves),
// so W loads are float4 (all 4 j values) and fully coalesced along m.
// ---------------------------------------------------------------------------
__global__ __launch_bounds__(256) void gop_wmma_kernel(
    const float* __restrict__ Wt,  // (bs, N, N, 4)
    const float* __restrict__ Xt,  // (bs, 32, N)
    float* __restrict__ Out,       // (bs, CH, N), writes channels [0,128)
    int N, int CH) {
  // Ws row pitch 132 floats (528B = 33*16, float4-aligned; lane stride
  // 132 mod 64 = 4 banks -> conflict-free for 16 lanes).
  __shared__ __align__(16) float Ws[16][132];  // [n][m*4 + j], m in [0,32)
  __shared__ __align__(16) float Xs[32][36];   // [f][m]

  const int tid = threadIdx.x;
  const int wv = tid >> 5;
  const int lane = tid & 31;
  const int half = lane >> 4;  // K-pair selector (ISA A/C layout)
  const int lid = lane & 15;
  const int j = wv >> 1;
  const int fb = (wv & 1) * 16;

  const int ntiles = N >> 4;
  const int b = blockIdx.x / ntiles;
  const int n0 = (blockIdx.x % ntiles) << 4;

  const size_t wbase = ((size_t)b * N + n0) * (size_t)N * 4;
  const size_t xbase = (size_t)b * 32 * (size_t)N;

  v8f acc = {};

  for (int m0 = 0; m0 < N; m0 += 32) {
    // Stage W tile: 16 n x 32 m x 4 j  = 512 float4, 2 per thread.
#pragma unroll
    for (int i = 0; i < 2; ++i) {
      const int idx = tid + i * 256;
      const int nn = idx >> 5, mm = idx & 31;
      const size_t goff = wbase + ((size_t)nn * N + (size_t)(m0 + mm)) * 4;
      *(float4*)&Ws[nn][mm * 4] = *(const float4*)(Wt + goff);
      // Prefetch next chunk (global_prefetch_b8 on gfx1250).
      if (m0 + 32 < N) __builtin_prefetch(Wt + goff + 128, 0, 1);
    }
    // Stage X tile: 32 f x 32 m = 256 float4, 1 per thread.
    {
      const int ff = tid >> 3, mq = tid & 7;
      *(float4*)&Xs[ff][mq * 4] =
          *(const float4*)(Xt + xbase + (size_t)ff * N + m0 + mq * 4);
    }
    __syncthreads();
#pragma unroll
    for (int k = 0; k < 32; k += 4) {
      const int ka = k + half * 2;
      v2f a, bb;
      a[0] = Xs[fb + lid][ka];
      a[1] = Xs[fb + lid][ka + 1];
      bb[0] = Ws[lid][ka * 4 + j];
      bb[1] = Ws[lid][(ka + 1) * 4 + j];
      acc = wmma16x16x4(a, bb, acc);
    }
    __syncthreads();
  }

#pragma unroll
  for (int v = 0; v < 8; ++v) {
    const int c = j * 32 + fb + v + 8 * half;
    Out[((size_t)b * CH + c) * N + n0 + lid] = acc[v];
  }
}

// ---------------------------------------------------------------------------
// pmul: Out[b, coff+f, n] = sum_m P[b, n, m] * XL[b, f, m]
// Tile: 64 n x 32 f per workgroup; P rows contiguous in m -> float4 staging.
// ---------------------------------------------------------------------------
__global__ __launch_bounds__(256) void pmul_wmma_kernel(
    const float* __restrict__ P,   // (bs, N, M)
    const float* __restrict__ XL,  // (bs, 32, M)
    float* __restrict__ Out,       // (bs, CH, N)
    int N, int M, int CH, int coff) {
  __shared__ __align__(16) float Ps[64][36];
  __shared__ __align__(16) float Xs[32][36];

  const int tid = threadIdx.x;
  const int wv = tid >> 5;
  const int lane = tid & 31;
  const int half = lane >> 4;
  const int lid = lane & 15;
  const int nt = wv >> 1;          // 0..3 -> n sub-tile
  const int fb = (wv & 1) * 16;

  const int nblk = N >> 6;
  const int b = blockIdx.x / nblk;
  const int n0 = (blockIdx.x % nblk) << 6;

  v8f acc = {};

  for (int m0 = 0; m0 < M; m0 += 32) {
#pragma unroll
    for (int i = 0; i < 2; ++i) {
      const int idx = tid + i * 256;
      const int nn = idx >> 3, mq = idx & 7;
      *(float4*)&Ps[nn][mq * 4] =
          *(const float4*)(P + ((size_t)b * N + n0 + nn) * M + m0 + mq * 4);
    }
    {
      const int ff = tid >> 3, mq = tid & 7;
      *(float4*)&Xs[ff][mq * 4] =
          *(const float4*)(XL + ((size_t)b * 32 + ff) * M + m0 + mq * 4);
    }
    __syncthreads();
#pragma unroll
    for (int k = 0; k < 32; k += 4) {
      const int ka = k + half * 2;
      v2f a, bb;
      a[0] = Xs[fb + lid][ka];
      a[1] = Xs[fb + lid][ka + 1];
      bb[0] = Ps[nt * 16 + lid][ka];
      bb[1] = Ps[nt * 16 + lid][ka + 1];
      acc = wmma16x16x4(a, bb, acc);
    }
    __syncthreads();
  }

#pragma unroll
  for (int v = 0; v < 8; ++v) {
    const int c = coff + fb + v + 8 * half;
    Out[((size_t)b * CH + c) * N + n0 + nt * 16 + lid] = acc[v];
  }
}

// ---------------------------------------------------------------------------
// pmulT: Out[b, coff+c, m] = sum_n P[b, n, m] * Z[b, c, n]   (c in [0,64))
// Tile: 64 c x 32 m; B operand (P) is lane-contiguous in m -> direct loads;
// A operand (Z) staged in LDS.
// ---------------------------------------------------------------------------
__global__ __launch_bounds__(256) void pmulT_wmma_kernel(
    const float* __restrict__ P,  // (bs, N, M)
    const float* __restrict__ Z,  // (bs, 64, N)
    float* __restrict__ Out,      // (bs, CH, M)
    int N, int M, int CH, int coff) {
  __shared__ __align__(16) float Zs[64][36];

  const int tid = threadIdx.x;
  const int wv = tid >> 5;
  const int lane = tid & 31;
  const int half = lane >> 4;
  const int lid = lane & 15;
  const int cb = (wv >> 1) * 16;  // 0,16,32,48
  const int mb = (wv & 1) * 16;

  const int mblk = M >> 5;
  const int b = blockIdx.x / mblk;
  const int m0 = (blockIdx.x % mblk) << 5;

  v8f acc = {};

  for (int n0 = 0; n0 < N; n0 += 32) {
#pragma unroll
    for (int i = 0; i < 2; ++i) {
      const int idx = tid + i * 256;
      const int cc = idx >> 3, nq = idx & 7;
      *(float4*)&Zs[cc][nq * 4] =
          *(const float4*)(Z + ((size_t)b * 64 + cc) * N + n0 + nq * 4);
    }
    __syncthreads();
#pragma unroll
    for (int k = 0; k < 32; k += 4) {
      const int ka = k + half * 2;
      v2f a, bb;
      a[0] = Zs[cb + lid][ka];
      a[1] = Zs[cb + lid][ka + 1];
      const float* prow =
          P + ((size_t)b * N + n0 + ka) * M + m0 + mb + lid;
      bb[0] = prow[0];
      bb[1] = prow[M];
      acc = wmma16x16x4(a, bb, acc);
    }
    __syncthreads();
  }

#pragma unroll
  for (int v = 0; v < 8; ++v) {
    const int c = coff + cb + v + 8 * half;
    Out[((size_t)b * CH + c) * M + m0 + mb + lid] = acc[v];
  }
}

// ---------------------------------------------------------------------------
// conv1x1 pair + bias + relu: Z[b,o,l]; o<32 -> wLin (linear), o>=32 -> wRe+ReLU
// ---------------------------------------------------------------------------
__global__ __launch_bounds__(256) void conv_wmma_kernel(
    const float* __restrict__ In,  // (bs, CHin, L)
    const float* __restrict__ wLin, const float* __restrict__ bLin,
    const float* __restrict__ wRe, const float* __restrict__ bRe,
    float* __restrict__ Zraw,  // (bs, 64, L)
    int CHin, int L) {
  __shared__ __align__(16) float Wk[64][36];
  __shared__ __align__(16) float Ins[32][36];

  const int tid = threadIdx.x;
  const int wv = tid >> 5;
  const int lane = tid & 31;
  const int half = lane >> 4;
  const int lid = lane & 15;
  const int ob = (wv >> 1) * 16;
  const int lb = (wv & 1) * 16;

  const int lblk = L >> 5;
  const int b = blockIdx.x / lblk;
  const int l0 = (blockIdx.x % lblk) << 5;

  v8f acc = {};

  for (int k0 = 0; k0 < CHin; k0 += 32) {
#pragma unroll
    for (int i = 0; i < 2; ++i) {
      const int idx = tid + i * 256;
      const int oo = idx >> 3, kq = idx & 7;
      const float* wsrc = (oo < 32) ? (wLin + (size_t)oo * CHin)
                                    : (wRe + (size_t)(oo - 32) * CHin);
      *(float4*)&Wk[oo][kq * 4] = *(const float4*)(wsrc + k0 + kq * 4);
    }
    {
      const int kk = tid >> 3, lq = tid & 7;
      *(float4*)&Ins[kk][lq * 4] =
          *(const float4*)(In + ((size_t)b * CHin + k0 + kk) * L + l0 + lq * 4);
    }
    __syncthreads();
#pragma unroll
    for (int k = 0; k < 32; k += 4) {
      const int ka = k + half * 2;
      v2f a, bb;
      a[0] = Wk[ob + lid][ka];
      a[1] = Wk[ob + lid][ka + 1];
      bb[0] = Ins[ka][lb + lid];
      bb[1] = Ins[ka + 1][lb + lid];
      acc = wmma16x16x4(a, bb, acc);
    }
    __syncthreads();
  }

#pragma unroll
  for (int v = 0; v < 8; ++v) {
    const int o = ob + v + 8 * half;
    const int l = l0 + lb + lid;
    const float bias = (o < 32) ? bLin[o] : bRe[o - 32];
    float val = acc[v] + bias;
    if (o >= 32) val = fmaxf(val, 0.0f);
    Zraw[((size_t)b * 64 + o) * L + l] = val;
  }
}

// ---------------------------------------------------------------------------
// masked BN stats: one block per channel (64 blocks)
// stats[c] = mean, stats[64+c] = 1/sqrt(var+eps)
// ---------------------------------------------------------------------------
__global__ __launch_bounds__(256) void bn_stats_kernel(
    const float* __restrict__ Z, const float* __restrict__ mask,
    const float* __restrict__ counts, int bs, int L,
    float* __restrict__ stats) {
  __shared__ float r1[256], r2[256];
  const int c = blockIdx.x;
  const int tid = threadIdx.x;
  float s1 = 0.f, s2 = 0.f;
  const int total = bs * L;
  for (int i = tid; i < total; i += 256) {
    const int b = i / L, l = i % L;
    const float m = mask[(size_t)b * L + l];
    const float z = Z[((size_t)b * 64 + c) * L + l];
    s1 += z * m;
    s2 += z * z * m;
  }
  r1[tid] = s1;
  r2[tid] = s2;
  __syncthreads();
  for (int s = 128; s > 0; s >>= 1) {
    if (tid < s) {
      r1[tid] += r1[tid + s];
      r2[tid] += r2[tid + s];
    }
    __syncthreads();
  }
  if (tid == 0) {
    float tot = 0.f;
    for (int b = 0; b < bs; ++b) tot += counts[b];
    const float mean = r1[0] / tot;
    const float var = r2[0] / tot - mean * mean;
    stats[c] = mean;
    stats[64 + c] = rsqrtf(var + 1e-5f);
  }
}

__global__ __launch_bounds__(256) void bn_apply_kernel(
    const float* __restrict__ Z, const float* __restrict__ stats,
    const float* __restrict__ mask, float* __restrict__ Out, int bs, int L) {
  const size_t i = (size_t)blockIdx.x * 256 + threadIdx.x;
  const size_t total = (size_t)bs * 64 * L;
  if (i >= total) return;
  const int l = (int)(i % L);
  const int c = (int)((i / L) % 64);
  const int b = (int)(i / ((size_t)64 * L));
  Out[i] = (Z[i] - stats[c]) * stats[64 + c] * mask[(size_t)b * L + l];
}

// ---------------------------------------------------------------------------
extern "C" void kernel_launch(void* const* d_in, const int* in_sizes, int n_in,
                              void* d_out, int out_size, void* d_ws,
                              size_t ws_size, hipStream_t stream) {
  const float* X = (const float*)d_in[0];
  const float* XL = (const float*)d_in[1];
  const float* W = (const float*)d_in[2];
  const float* WL = (const float*)d_in[3];
  const float* Pm = (const float*)d_in[4];
  const float* Pd = (const float*)d_in[5];
  const float* Nb = (const float*)d_in[6];
  const float* mask = (const float*)d_in[7];
  const float* Eb = (const float*)d_in[8];
  const float* mask_lg = (const float*)d_in[9];
  const float* w1 = (const float*)d_in[10];
  const float* b1 = (const float*)d_in[11];
  const float* w2 = (const float*)d_in[12];
  const float* b2 = (const float*)d_in[13];
  const float* w3 = (const float*)d_in[14];
  const float* b3 = (const float*)d_in[15];
  const float* w4 = (const float*)d_in[16];
  const float* b4 = (const float*)d_in[17];

  const int bs = in_sizes[6];           // 8
  const int N = in_sizes[7] / bs;       // 1024
  const int M = in_sizes[9] / bs;       // 2048
  const int CH1 = 192;                  // J*F + 2F
  const int CH2 = 256;                  // J*F + 4*n_out

  // workspace layout (floats)
  float* ws = (float*)d_ws;
  float* x1 = ws;                                  // bs*192*N
  float* xd1 = x1 + (size_t)bs * CH1 * N;          // bs*256*M
  float* zb = xd1 + (size_t)bs * CH2 * M;          // bs*64*N
  float* zdb = zb + (size_t)bs * 64 * N;           // bs*64*M
  float* st1 = zdb + (size_t)bs * 64 * M;          // 128
  float* st2 = st1 + 128;                          // 128

  // d_out layout: [zbn1, zdbn1, W, WL, Pm, Pd]
  float* out = (float*)d_out;
  float* o_zbn = out;
  float* o_zdbn = o_zbn + (size_t)bs * 64 * N;
  float* o_W = o_zdbn + (size_t)bs * 64 * M;
  float* o_WL = o_W + (size_t)in_sizes[2];
  float* o_Pm = o_WL + (size_t)in_sizes[3];
  float* o_Pd = o_Pm + (size_t)in_sizes[4];

  // ---- stage 1: x1 = [gop(W,X) | Pm*XL | Pd*XL] -------------------------
  gop_wmma_kernel<<<bs * (N / 16), 256, 0, stream>>>(W, X, x1, N, CH1);
  pmul_wmma_kernel<<<bs * (N / 64), 256, 0, stream>>>(Pm, XL, x1, N, M, CH1,
                                                      128);
  pmul_wmma_kernel<<<bs * (N / 64), 256, 0, stream>>>(Pd, XL, x1, N, M, CH1,
                                                      160);
  // zb = [conv(w2)+b2 | relu(conv(w1)+b1)]
  conv_wmma_kernel<<<bs * (N / 32), 256, 0, stream>>>(x1, w2, b2, w1, b1, zb,
                                                      CH1, N);
  bn_stats_kernel<<<64, 256, 0, stream>>>(zb, mask, Nb, bs, N, st1);
  {
    const size_t tot = (size_t)bs * 64 * N;
    bn_apply_kernel<<<(unsigned)((tot + 255) / 256), 256, 0, stream>>>(
        zb, st1, mask, o_zbn, bs, N);
  }

  // ---- stage 2: xd1 = [gop(WL,XL) | PmT*zbn1 | PdT*zbn1] -----------------
  gop_wmma_kernel<<<bs * (M / 16), 256, 0, stream>>>(WL, XL, xd1, M, CH2);
  pmulT_wmma_kernel<<<bs * (M / 32), 256, 0, stream>>>(Pm, o_zbn, xd1, N, M,
                                                       CH2, 128);
  pmulT_wmma_kernel<<<bs * (M / 32), 256, 0, stream>>>(Pd, o_zbn, xd1, N, M,
                                                       CH2, 192);
  conv_wmma_kernel<<<bs * (M / 32), 256, 0, stream>>>(xd1, w4, b4, w3, b3, zdb,
                                                      CH2, M);
  bn_stats_kernel<<<64, 256, 0, stream>>>(zdb, mask_lg, Eb, bs, M, st2);
  {
    const size_t tot = (size_t)bs * 64 * M;
    bn_apply_kernel<<<(unsigned)((tot + 255) / 256), 256, 0, stream>>>(
        zdb, st2, mask_lg, o_zdbn, bs, M);
  }

  // ---- passthrough outputs (W, WL, Pm, Pd) -------------------------------
  (void)hipMemcpyAsync(o_W, W, (size_t)in_sizes[2] * sizeof(float),
                       hipMemcpyDeviceToDevice, stream);
  (void)hipMemcpyAsync(o_WL, WL, (size_t)in_sizes[3] * sizeof(float),
                       hipMemcpyDeviceToDevice, stream);
  (void)hipMemcpyAsync(o_Pm, Pm, (size_t)in_sizes[4] * sizeof(float),
                       hipMemcpyDeviceToDevice, stream);
  (void)hipMemcpyAsync(o_Pd, Pd, (size_t)in_sizes[5] * sizeof(float),
                       hipMemcpyDeviceToDevice, stream);
}